// RipsLayer_64424509440837
// MI455X (gfx1250) — compile-verified
//
#include <hip/hip_runtime.h>
#include <math.h>

// Problem constants (reference): N=8192 points, D=3.
#define NPTS 8192
#define XFLOATS (NPTS * 3)          // 24576 floats
#define XBYTES  (XFLOATS * 4)       // 98304 bytes = 96 KB (fits 320 KB/WGP LDS)
#define XCHUNKS (XBYTES / 16)       // 6144 b128 chunks

// ---------------------------------------------------------------------------
// Staged kernel: async-copy X into LDS (CDNA5 ASYNCcnt path), then do all
// index-pair distance gathers out of LDS. One unified grid-stride task loop
// covers every output element so the poisoned d_out is fully overwritten.
//
// Output layout (flat, return order):
//   [0, 2*F0)              finite_dgm0  rows (0, death)
//   [2*F0, 2*F0+E0)        essential_dgm0 zeros
//   [base1, base1+2*F1)    finite_dgm1 flattened distances
//   [base2, base2+E1)      essential_dgm1 distances
// ---------------------------------------------------------------------------
__global__ __launch_bounds__(256) void rips_staged_kernel(
    const float* __restrict__ X,
    const int*   __restrict__ idx0f,   // [F0,3]
    const int*   __restrict__ idx1f,   // [F1,4]
    const int*   __restrict__ idx1e,   // [E1,2]
    float*       __restrict__ out,
    int f0, int e0, int f1, int e1)
{
    __shared__ float sX[XFLOATS];

    // ---- Stage X into LDS via CDNA5 async global->LDS b128 copies ----------
    {
        // Low 32 bits of a generic pointer to LDS == byte offset within the
        // workgroup's LDS allocation.
        const unsigned ldsBase = (unsigned)(unsigned long long)(&sX[0]);
        for (int c = threadIdx.x; c < XCHUNKS; c += blockDim.x) {
            unsigned goff  = (unsigned)c * 16u;      // VADDR: 32-bit byte offset (GVS mode)
            unsigned laddr = ldsBase + goff;          // VDST : LDS byte address
            asm volatile("global_load_async_to_lds_b128 %0, %1, %2"
                         :: "v"(laddr), "v"(goff), "s"(X)
                         : "memory");
        }
        // Drain this wave's ASYNCcnt, then barrier so every wave sees all of sX.
        asm volatile("s_wait_asynccnt 0x0" ::: "memory");
    }
    __syncthreads();

    auto dist = [&](int a, int b) -> float {
        float dx = sX[a * 3 + 0] - sX[b * 3 + 0];
        float dy = sX[a * 3 + 1] - sX[b * 3 + 1];
        float dz = sX[a * 3 + 2] - sX[b * 3 + 2];
        return sqrtf(dx * dx + dy * dy + dz * dz);
    };

    const int base1 = 2 * f0 + e0;
    const int base2 = base1 + 2 * f1;
    const int T     = f0 + e0 + 2 * f1 + e1;   // one task per distance/zero

    for (int t = blockIdx.x * blockDim.x + threadIdx.x; t < T;
         t += gridDim.x * blockDim.x) {
        if (t < f0) {
            // dim-0 finite pair: (vertex, edge_v1, edge_v2) -> (0, |edge|)
            int a = idx0f[3 * t + 1];
            int b = idx0f[3 * t + 2];
            out[2 * t + 0] = 0.0f;
            out[2 * t + 1] = dist(a, b);
        } else if (t < f0 + e0) {
            out[2 * f0 + (t - f0)] = 0.0f;     // essential dim-0 births
        } else if (t < f0 + e0 + 2 * f1) {
            int k = t - (f0 + e0);             // flattened [F1,2] gather
            int a = idx1f[2 * k + 0];
            int b = idx1f[2 * k + 1];
            out[base1 + k] = dist(a, b);
        } else {
            int e = t - (f0 + e0 + 2 * f1);
            int a = idx1e[2 * e + 0];
            int b = idx1e[2 * e + 1];
            out[base2 + e] = dist(a, b);
        }
    }
}

// ---------------------------------------------------------------------------
// Fallback: direct global gathers (used only if N != 8192).
// ---------------------------------------------------------------------------
__global__ __launch_bounds__(256) void rips_direct_kernel(
    const float* __restrict__ X,
    const int*   __restrict__ idx0f,
    const int*   __restrict__ idx1f,
    const int*   __restrict__ idx1e,
    float*       __restrict__ out,
    int f0, int e0, int f1, int e1)
{
    auto dist = [&](int a, int b) -> float {
        float dx = X[a * 3 + 0] - X[b * 3 + 0];
        float dy = X[a * 3 + 1] - X[b * 3 + 1];
        float dz = X[a * 3 + 2] - X[b * 3 + 2];
        return sqrtf(dx * dx + dy * dy + dz * dz);
    };

    const int base1 = 2 * f0 + e0;
    const int base2 = base1 + 2 * f1;
    const int T     = f0 + e0 + 2 * f1 + e1;

    for (int t = blockIdx.x * blockDim.x + threadIdx.x; t < T;
         t += gridDim.x * blockDim.x) {
        if (t < f0) {
            int a = idx0f[3 * t + 1];
            int b = idx0f[3 * t + 2];
            out[2 * t + 0] = 0.0f;
            out[2 * t + 1] = dist(a, b);
        } else if (t < f0 + e0) {
            out[2 * f0 + (t - f0)] = 0.0f;
        } else if (t < f0 + e0 + 2 * f1) {
            int k = t - (f0 + e0);
            int a = idx1f[2 * k + 0];
            int b = idx1f[2 * k + 1];
            out[base1 + k] = dist(a, b);
        } else {
            int e = t - (f0 + e0 + 2 * f1);
            int a = idx1e[2 * e + 0];
            int b = idx1e[2 * e + 1];
            out[base2 + e] = dist(a, b);
        }
    }
}

extern "C" void kernel_launch(void* const* d_in, const int* in_sizes, int n_in,
                              void* d_out, int out_size, void* d_ws, size_t ws_size,
                              hipStream_t stream) {
    const float* X     = (const float*)d_in[0];
    const int*   idx0f = (const int*)d_in[1];
    // d_in[2] = idx0_essential: values unused (births are all zero); only count matters.
    const int*   idx1f = (const int*)d_in[3];
    const int*   idx1e = (const int*)d_in[4];

    const int n  = in_sizes[0] / 3;   // points
    const int f0 = in_sizes[1] / 3;   // dim-0 finite triples
    const int e0 = in_sizes[2];       // dim-0 essential classes
    const int f1 = in_sizes[3] / 4;   // dim-1 finite quadruples
    const int e1 = in_sizes[4] / 2;   // dim-1 essential pairs

    float* out = (float*)d_out;

    const dim3 block(256);            // 8 wave32s per workgroup
    if (n == NPTS) {
        const dim3 grid(16);          // ~1 WG per WGP; ~4 tasks/thread
        rips_staged_kernel<<<grid, block, 0, stream>>>(
            X, idx0f, idx1f, idx1e, out, f0, e0, f1, e1);
    } else {
        const int T = f0 + e0 + 2 * f1 + e1;
        const int g = (T + 255) / 256 < 64 ? (T + 255) / 256 : 64;
        rips_direct_kernel<<<dim3(g > 0 ? g : 1), block, 0, stream>>>(
            X, idx0f, idx1f, idx1e, out, f0, e0, f1, e1);
    }
}